// GATv2Encoder_8315056685617
// MI455X (gfx1250) — compile-verified
//
#include <hip/hip_runtime.h>
#include <hip/hip_bf16.h>

#define DIM 128

typedef float v2f __attribute__((ext_vector_type(2)));
typedef float v8f __attribute__((ext_vector_type(8)));

// ---------------------------------------------------------------------------
// GEMM: Out[N x 128] = H[N x 128] @ W[128 x 128] using V_WMMA_F32_16X16X4_F32.
// One wave (32 lanes) computes a 16x128 strip: 8 accumulators of 16x16 f32.
// W is staged in LDS (64KB) once per block; A rows streamed from global.
// ---------------------------------------------------------------------------
__global__ void __launch_bounds__(256) gcn_gemm_wmma(
    const float* __restrict__ H, const float* __restrict__ W,
    float* __restrict__ Out, int nrows) {
  __shared__ float sW[DIM * DIM];  // 64 KB of the 320 KB/WGP LDS
  const int tid = threadIdx.x;
  for (int i = tid; i < DIM * DIM; i += 256) sW[i] = W[i];
  __syncthreads();

  const int wave = tid >> 5;
  const int lane = tid & 31;
  const int m0 = (blockIdx.x * 8 + wave) * 16;
  if (m0 >= nrows) return;           // uniform per-wave: EXEC stays all-ones

  const int half = lane >> 4;        // 0: lanes 0-15, 1: lanes 16-31
  const int l16  = lane & 15;
  const int mrow = m0 + l16;
  const float* __restrict__ arow = H + (size_t)(mrow < nrows ? mrow : nrows - 1) * DIM;

  v8f acc[8] = {};                   // 8 N-tiles of 16x16 f32 (64 VGPRs)

  for (int k = 0; k < DIM; k += 4) {
    // A frag (16x4): lane-half selects K pair {k,k+1} vs {k+2,k+3}
    v2f a;
    a.x = arow[k + half * 2 + 0];
    a.y = arow[k + half * 2 + 1];
#pragma unroll
    for (int t = 0; t < 8; ++t) {
      const int n = t * 16 + l16;
      v2f b;                         // B frag (4x16), K striped by lane half
      b.x = sW[(k + half * 2 + 0) * DIM + n];
      b.y = sW[(k + half * 2 + 1) * DIM + n];
      acc[t] = __builtin_amdgcn_wmma_f32_16x16x4_f32(
          /*neg_a=*/false, a, /*neg_b=*/false, b,
          /*c_mod=*/(short)0, acc[t], /*reuse_a=*/false, /*reuse_b=*/false);
    }
  }

  // C/D layout: VGPR v -> M = v (lanes 0-15) / v+8 (lanes 16-31), N = l16.
  // Wave-uniform full-tile check: keeps the hot path free of per-lane EXEC
  // juggling (nrows % 16 == 0 in this workload, so this is always taken).
  float* __restrict__ obase = Out + (size_t)(m0 + half * 8) * DIM + l16;
  if (m0 + 16 <= nrows) {
#pragma unroll
    for (int v = 0; v < 8; ++v) {
#pragma unroll
      for (int t = 0; t < 8; ++t) {
        obase[(size_t)v * DIM + t * 16] = acc[t][v];
      }
    }
  } else {
#pragma unroll
    for (int v = 0; v < 8; ++v) {
      if (m0 + v + half * 8 < nrows) {
#pragma unroll
        for (int t = 0; t < 8; ++t) {
          obase[(size_t)v * DIM + t * 16] = acc[t][v];
        }
      }
    }
  }
}

// ---------------------------------------------------------------------------
// Degree / normalization precompute (reused by all 5 layers)
// ---------------------------------------------------------------------------
__global__ void fill_zero(float* __restrict__ p, long n) {
  long i = (long)blockIdx.x * blockDim.x + threadIdx.x;
  if (i < n) p[i] = 0.0f;
}

__global__ void deg_accum(const int* __restrict__ col, const float* __restrict__ ew,
                          float* __restrict__ deg, int E) {
  int e = blockIdx.x * blockDim.x + threadIdx.x;
  if (e < E) atomicAdd(&deg[col[e]], ew[e]);
}

__global__ void deg_to_dinv(float* __restrict__ deg, int n) {
  int i = blockIdx.x * blockDim.x + threadIdx.x;
  if (i < n) deg[i] = rsqrtf(deg[i] + 1.0f);  // +1 = self-loop weight; deg>0
}

__global__ void edge_norm(const int* __restrict__ row, const int* __restrict__ col,
                          const float* __restrict__ ew, const float* __restrict__ dinv,
                          float* __restrict__ nrm, int E) {
  int e = blockIdx.x * blockDim.x + threadIdx.x;
  if (e < E) nrm[e] = dinv[row[e]] * ew[e] * dinv[col[e]];
}

// ---------------------------------------------------------------------------
// Aggregation: B = bias + selfloop (non-atomic init), then edge scatter atomics
// ---------------------------------------------------------------------------
__global__ void init_bias_selfloop(const float* __restrict__ A,
                                   const float* __restrict__ dinv,
                                   const float* __restrict__ bias,
                                   float* __restrict__ B, int n) {
  long i = (long)blockIdx.x * blockDim.x + threadIdx.x;
  if (i >= (long)n * DIM) return;
  int node = (int)(i >> 7);
  int f = (int)(i & (DIM - 1));
  float di = dinv[node];
  B[i] = bias[f] + A[i] * (di * di);
}

__global__ void __launch_bounds__(256) scatter_edges(
    const float* __restrict__ A, const int* __restrict__ row,
    const int* __restrict__ col, const float* __restrict__ nrm,
    float* __restrict__ B, int E) {
  int e = blockIdx.x * 8 + (threadIdx.x >> 5);   // one wave32 per edge
  if (e >= E) return;
  int lane = threadIdx.x & 31;                   // 4 floats per lane = 128
  int r = row[e], c = col[e];
  float w = nrm[e];
  float4 v = ((const float4*)(A + (size_t)r * DIM))[lane];
  float* dst = B + (size_t)c * DIM + lane * 4;
  atomicAdd(dst + 0, v.x * w);
  atomicAdd(dst + 1, v.y * w);
  atomicAdd(dst + 2, v.z * w);
  atomicAdd(dst + 3, v.w * w);
}

// ---------------------------------------------------------------------------
// BatchNorm (training): per-feature sum / sumsq, then fused scale+shift+ReLU
// ---------------------------------------------------------------------------
__global__ void __launch_bounds__(256) bn_stats(const float* __restrict__ B,
                                                float* __restrict__ sums, int n) {
  __shared__ float s0[256];
  __shared__ float s1[256];
  int tid = threadIdx.x;
  int f = tid & (DIM - 1);
  int slot = tid >> 7;  // 2 threads per feature
  float sum = 0.0f, sq = 0.0f;
  int start = blockIdx.x * 256;
  int end = min(n, start + 256);
  for (int node = start + slot; node < end; node += 2) {
    float v = B[(size_t)node * DIM + f];
    sum += v;
    sq += v * v;
  }
  s0[tid] = sum;
  s1[tid] = sq;
  __syncthreads();
  if (slot == 0) {
    atomicAdd(&sums[f], s0[tid] + s0[tid + 128]);
    atomicAdd(&sums[DIM + f], s1[tid] + s1[tid + 128]);
  }
}

__global__ void bn_apply(const float* __restrict__ sums,
                         const float* __restrict__ gamma,
                         const float* __restrict__ beta,
                         const float* __restrict__ Bin, float* __restrict__ Hout,
                         int n, int relu) {
  long i = (long)blockIdx.x * blockDim.x + threadIdx.x;
  if (i >= (long)n * DIM) return;
  int f = (int)(i & (DIM - 1));
  float inv_n = 1.0f / (float)n;
  float mean = sums[f] * inv_n;
  float var = sums[DIM + f] * inv_n - mean * mean;  // biased, matches BN train
  float scale = gamma[f] * rsqrtf(var + 1e-5f);
  float shift = beta[f] - mean * scale;
  float v = Bin[i] * scale + shift;
  if (relu) v = fmaxf(v, 0.0f);
  Hout[i] = v;
}

// ---------------------------------------------------------------------------
// global_add_pool: one wave per node, float4 per lane, atomic into xpool
// ---------------------------------------------------------------------------
__global__ void __launch_bounds__(256) pool_nodes(const float* __restrict__ H,
                                                  const int* __restrict__ batch,
                                                  float* __restrict__ xpool, int n) {
  int node = blockIdx.x * 8 + (threadIdx.x >> 5);
  if (node >= n) return;
  int lane = threadIdx.x & 31;
  int g = batch[node];
  float4 v = ((const float4*)(H + (size_t)node * DIM))[lane];
  float* dst = xpool + (size_t)g * DIM + lane * 4;
  atomicAdd(dst + 0, v.x);
  atomicAdd(dst + 1, v.y);
  atomicAdd(dst + 2, v.z);
  atomicAdd(dst + 3, v.w);
}

// ---------------------------------------------------------------------------
extern "C" void kernel_launch(void* const* d_in, const int* in_sizes, int n_in,
                              void* d_out, int out_size, void* d_ws, size_t ws_size,
                              hipStream_t stream) {
  const int*   batch  = (const int*)d_in[0];
  const float* x      = (const float*)d_in[1];
  const int*   eidx   = (const int*)d_in[2];
  const float* ew     = (const float*)d_in[3];
  const float* Ws     = (const float*)d_in[4];
  const float* bs     = (const float*)d_in[5];
  const float* gammas = (const float*)d_in[6];
  const float* betas  = (const float*)d_in[7];

  const int N = in_sizes[0];
  const int E = in_sizes[3];
  const int* row = eidx;          // edge_index[0]
  const int* col = eidx + E;      // edge_index[1]

  float* out   = (float*)d_out;
  const int G  = (out_size - N * DIM) / DIM;  // number of graphs (256)
  float* xpool = out;                         // [G, 128]
  float* hout  = out + (size_t)G * DIM;       // [N, 128]

  // Workspace layout (floats): A | B | dinv | edge_norm | bn_sums(256)
  float* A    = (float*)d_ws;
  float* B    = A + (size_t)N * DIM;
  float* dinv = B + (size_t)N * DIM;
  float* nrm  = dinv + N;
  float* sums = nrm + E;

  const long NF = (long)N * DIM;

  // --- precompute symmetric-normalization coefficients (reused 5x) ---
  fill_zero<<<(N + 255) / 256, 256, 0, stream>>>(dinv, N);
  deg_accum<<<(E + 255) / 256, 256, 0, stream>>>(col, ew, dinv, E);
  deg_to_dinv<<<(N + 255) / 256, 256, 0, stream>>>(dinv, N);
  edge_norm<<<(E + 255) / 256, 256, 0, stream>>>(row, col, ew, dinv, nrm, E);

  const float* hin = x;
  for (int i = 0; i < 5; ++i) {
    const int gemm_blocks = ((N + 15) / 16 + 7) / 8;
    gcn_gemm_wmma<<<gemm_blocks, 256, 0, stream>>>(hin, Ws + (size_t)i * DIM * DIM, A, N);
    init_bias_selfloop<<<(int)((NF + 255) / 256), 256, 0, stream>>>(A, dinv, bs + i * DIM, B, N);
    scatter_edges<<<(E + 7) / 8, 256, 0, stream>>>(A, row, col, nrm, B, E);
    fill_zero<<<1, 256, 0, stream>>>(sums, 2 * DIM);
    bn_stats<<<(N + 255) / 256, 256, 0, stream>>>(B, sums, N);
    float* ho = (i == 4) ? hout : B;  // in-place BN except last layer -> d_out
    bn_apply<<<(int)((NF + 255) / 256), 256, 0, stream>>>(
        sums, gammas + i * DIM, betas + i * DIM, B, ho, N, (i < 4) ? 1 : 0);
    hin = ho;
  }

  fill_zero<<<(G * DIM + 255) / 256, 256, 0, stream>>>(xpool, (long)G * DIM);
  pool_nodes<<<(N + 7) / 8, 256, 0, stream>>>(hout, batch, xpool, N);
}